// HungarianMatcher_28406913696524
// MI455X (gfx1250) — compile-verified
//
#include <hip/hip_runtime.h>

typedef __attribute__((ext_vector_type(2))) float v2f;
typedef __attribute__((ext_vector_type(8))) float v8f;

#define NC 92          // classes (K dim), 92 = 23 * 4 -> exact K=4 tiling
#define QT 32          // queries per block (2 WMMA tiles of 16 in M)
#define TB 320         // targets per block (5 groups of 64 = 5 x (4 WMMA tiles of 16))
#define NTGT 1600      // total targets (output row length)

// C[q,t] = 5*L1 + 1*(-prob[q,id[t]]) + 2*(-giou)
// cost_class done as exact one-hot GEMM via V_WMMA_F32_16X16X4_F32.
__launch_bounds__(256, 2)
__global__ void matcher_cost_kernel(const float* __restrict__ logits,   // [16000, 92]
                                    const float* __restrict__ qboxes,   // [16000, 4] cxcywh
                                    const int*   __restrict__ tids,     // [1600]
                                    const float* __restrict__ tboxes,   // [1600, 4] cxcywh
                                    float* __restrict__ out)            // [16000, 1600]
{
    __shared__ float  probs[QT][NC];           // staged logits, overwritten with softmax probs
    __shared__ float4 qraw[QT], qxy[QT];
    __shared__ float  qarea[QT];
    __shared__ float4 traw[TB], txy[TB];
    __shared__ float  tarea[TB];
    __shared__ int    tid_s[TB];

    const int tidx  = threadIdx.x;
    const int qbase = blockIdx.x * QT;
    const int tbase = blockIdx.y * TB;

    // Target tables are tiny and reused by every q-block -> keep hot (global_prefetch_b8).
    __builtin_prefetch(tboxes + (size_t)tbase * 4, 0, 3);
    __builtin_prefetch(tids + tbase, 0, 3);

    // ---- stage logits rows [qbase, qbase+QT) into LDS (coalesced) ----
    for (int i = tidx; i < QT * NC; i += 256) {
        probs[i / NC][i % NC] = logits[(size_t)qbase * NC + i];
    }
    // ---- stage + preprocess query boxes ----
    if (tidx < QT) {
        float4 b = ((const float4*)qboxes)[qbase + tidx];
        qraw[tidx] = b;
        float4 xy = make_float4(b.x - 0.5f * b.z, b.y - 0.5f * b.w,
                                b.x + 0.5f * b.z, b.y + 0.5f * b.w);
        qxy[tidx]   = xy;
        qarea[tidx] = (xy.z - xy.x) * (xy.w - xy.y);
    }
    // ---- stage + preprocess ALL of this block's target boxes / ids ----
    for (int t = tidx; t < TB; t += 256) {
        float4 b = ((const float4*)tboxes)[tbase + t];
        traw[t] = b;
        float4 xy = make_float4(b.x - 0.5f * b.z, b.y - 0.5f * b.w,
                                b.x + 0.5f * b.z, b.y + 0.5f * b.w);
        txy[t]   = xy;
        tarea[t] = (xy.z - xy.x) * (xy.w - xy.y);
        tid_s[t] = tids[tbase + t];
    }
    __syncthreads();

    // ---- softmax in place: one thread per query; exp evaluated once per element ----
    if (tidx < QT) {
        float m = -INFINITY;
        for (int k = 0; k < NC; ++k) m = fmaxf(m, probs[tidx][k]);
        float s = 0.0f;
        for (int k = 0; k < NC; ++k) {
            float e = __expf(probs[tidx][k] - m);
            probs[tidx][k] = e;
            s += e;
        }
        float inv = 1.0f / s;
        for (int k = 0; k < NC; ++k) probs[tidx][k] *= inv;
    }
    __syncthreads();

    // ---- per-wave 16x16 tiles: 8 waves cover 2(q) x 4(t) subtiles, loop over 5 t-groups ----
    const int wave = tidx >> 5;
    const int lane = tidx & 31;
    const int qsub = wave >> 2;         // 0..1
    const int tsub = wave & 3;          // 0..3
    const int ln16 = lane & 15;
    const int hi   = lane >> 4;         // 0|1
    const int khi  = hi * 2;            // f32 A/B layout: K offset per half-wave

    const int qloc0 = qsub * 16;
    const float* aptr = &probs[qloc0 + ln16][khi];  // A fragment base (lane-fixed)

    #pragma unroll 1
    for (int g = 0; g < 5; ++g) {
        const int tloc = g * 64 + tsub * 16 + ln16; // N index (same for both half-waves)
        const int myid = tid_s[tloc];

        // cost_class via exact one-hot WMMA: acc[r] = prob[qb+r+8*hi][id[t]]
        v8f acc = {0.f, 0.f, 0.f, 0.f, 0.f, 0.f, 0.f, 0.f};
        for (int k0 = 0; k0 < NC; k0 += 4) {
            // wave-uniform skip of K-blocks with no matching target id
            int inblk = (myid >= k0) && (myid < k0 + 4);
            if (__builtin_amdgcn_ballot_w32(inblk)) {
                // A (16x4 f32): lane L<16 -> M=L, K=k0..k0+1; L>=16 -> M=L-16, K=k0+2..k0+3
                v2f A; A.x = aptr[k0]; A.y = aptr[k0 + 1];
                // B (4x16 one-hot): N over lanes, K mirrors A
                v2f B;
                B.x = (myid == k0 + khi)     ? 1.0f : 0.0f;
                B.y = (myid == k0 + khi + 1) ? 1.0f : 0.0f;
                acc = __builtin_amdgcn_wmma_f32_16x16x4_f32(
                    false, A, false, B, (short)0, acc, false, false);
            }
        }

        // ---- epilogue: L1 + GIoU, fused combine, coalesced stores ----
        const float4 tr = traw[tloc];
        const float4 te = txy[tloc];
        const float  ta = tarea[tloc];

        #pragma unroll
        for (int r = 0; r < 8; ++r) {
            const int    ql = qloc0 + r + 8 * hi;
            const float4 qr = qraw[ql];
            const float4 qe = qxy[ql];
            const float  qa = qarea[ql];

            float l1 = fabsf(qr.x - tr.x) + fabsf(qr.y - tr.y) +
                       fabsf(qr.z - tr.z) + fabsf(qr.w - tr.w);

            float iw    = fminf(qe.z, te.z) - fmaxf(qe.x, te.x);
            float ih    = fminf(qe.w, te.w) - fmaxf(qe.y, te.y);
            float inter = fmaxf(iw, 0.0f) * fmaxf(ih, 0.0f);
            float uni   = qa + ta - inter;
            float iou   = inter * __builtin_amdgcn_rcpf(uni + 1e-6f);
            float ew    = fmaxf(qe.z, te.z) - fminf(qe.x, te.x);
            float eh    = fmaxf(qe.w, te.w) - fminf(qe.y, te.y);
            float enc   = ew * eh;
            float giou  = iou - (enc - uni) * __builtin_amdgcn_rcpf(enc + 1e-6f);

            float c = 5.0f * l1 - acc[r] - 2.0f * giou;
            out[(size_t)(qbase + ql) * NTGT + (tbase + tloc)] = c;
        }
    }
}

extern "C" void kernel_launch(void* const* d_in, const int* in_sizes, int n_in,
                              void* d_out, int out_size, void* d_ws, size_t ws_size,
                              hipStream_t stream) {
    const float* logits = (const float*)d_in[0];   // [16,1000,92] f32
    const float* pboxes = (const float*)d_in[1];   // [16,1000,4]  f32
    const int*   tids   = (const int*)  d_in[2];   // [1600]       i32
    const float* tboxes = (const float*)d_in[3];   // [1600,4]     f32
    float*       out    = (float*)d_out;           // [16,1000,1600] f32

    const int totalQ = in_sizes[1] / 4;            // 16000
    const int totalT = in_sizes[2];                // 1600

    dim3 grid(totalQ / QT, totalT / TB);           // (500, 5)
    matcher_cost_kernel<<<grid, 256, 0, stream>>>(logits, pboxes, tids, tboxes, out);
}